// RPNLoss_23381801960268
// MI455X (gfx1250) — compile-verified
//
#include <hip/hip_runtime.h>
#include <hip/hip_bf16.h>
#include <math.h>

// ---------------- constants ----------------
#define NBLK 512
#define TPB  256
#define GNUM 64

__device__ __constant__ float C_IOU_POS   = 0.8f;
__device__ __constant__ float C_IOU_NEG   = 0.3f;
__device__ __constant__ float C_LAMBDA    = 10.0f;
__device__ __constant__ float C_LOG1PEXP  = 0.31326168751822286f; // log1p(exp(-1))
__device__ __constant__ float C_LN2       = 0.6931471805599453f;  // log(2)

typedef float v2f __attribute__((ext_vector_type(2)));
typedef float v8f __attribute__((ext_vector_type(8)));

// ---------------- WMMA f32 16x16x4 helper ----------------
__device__ __forceinline__ v8f wmma4(v2f a, v2f b, v8f c) {
  // 8 args: (neg_a, A, neg_b, B, c_mod, C, reuse_a, reuse_b)
  return __builtin_amdgcn_wmma_f32_16x16x4_f32(false, a, false, b, (short)0, c,
                                               false, false);
}

// Exact wave32 reduction of 4 f32 accumulators using 6 chained
// v_wmma_f32_16x16x4_f32 ops.  On return, lane n (n=0..3) holds the wave
// total of q_n.  A-layout: lane m -> A[m][0..1], lane m+16 -> A[m][2..3].
// B-layout: VGPR0 = rows K0 (lanes 0-15) / K1 (lanes 16-31), VGPR1 = K2/K3.
__device__ __forceinline__ float wave_reduce4(int lane, float q0, float q1,
                                              float q2, float q3) {
  // Step 1: fold lane L with L+16, routing q0/q1 -> cols 0/1, q2/q3 -> cols 2/3
  float sel01 = (lane == 0 || lane == 17) ? 1.0f : 0.0f;
  float sel23 = (lane == 2 || lane == 19) ? 1.0f : 0.0f;
  v2f a01; a01.x = q0;    a01.y = q1;
  v2f a23; a23.x = q2;    a23.y = q3;
  v2f b01; b01.x = sel01; b01.y = sel01;
  v2f b23; b23.x = sel23; b23.y = sel23;
  v8f c = {};
  c = wmma4(a01, b01, c);   // cols 0,1 <- q0,q1 lane-half folds
  c = wmma4(a23, b23, c);   // cols 2,3 <- q2,q3 lane-half folds
  // Step 2: ones(16x4) x (rows of c) summed over all 16 rows, C-chained.
  v2f ones; ones.x = 1.0f; ones.y = 1.0f;
  v8f d = {};
  v2f b;
  b.x = c[0]; b.y = c[1]; d = wmma4(ones, b, d); // rows 0,8,1,9
  b.x = c[2]; b.y = c[3]; d = wmma4(ones, b, d); // rows 2,10,3,11
  b.x = c[4]; b.y = c[5]; d = wmma4(ones, b, d); // rows 4,12,5,13
  b.x = c[6]; b.y = c[7]; d = wmma4(ones, b, d); // rows 6,14,7,15
  return d[0];              // column n (lane n) = wave total of q_n
}

__device__ __forceinline__ float sl1f(float d) {
  float ad = fabsf(d);
  return ad < 1.0f ? 0.5f * d * d : ad - 0.5f;
}

// ---------------- init ----------------
__global__ void rpn_init(unsigned long long* best_key, int* cnts) {
  int i = threadIdx.x;
  if (i < GNUM) best_key[i] = 0ull;
  if (i < 2)    cnts[i] = 0;
}

// ---------------- pass 1: main sweep ----------------
__global__ __launch_bounds__(TPB) void rpn_main(
    const float* __restrict__ scores, const float* __restrict__ regions,
    const float* __restrict__ anchors, const float* __restrict__ gt,
    unsigned long long* __restrict__ best_key, int* __restrict__ cnts,
    float* __restrict__ cls_partial, float* __restrict__ reg_partial, int A) {
  const int t    = threadIdx.x;
  const int g    = t & 63;   // gt row owned by this thread
  const int sub  = t >> 6;   // 0..3 : anchor interleave within the chunk
  const int lane = t & 31;
  const int wave = t >> 5;   // 0..7

  // per-g constants (registers)
  const float t0 = gt[4 * g + 0], t1 = gt[4 * g + 1];
  const float t2 = gt[4 * g + 2], t3 = gt[4 * g + 3];
  const float area_g = (t2 - t0) * (t3 - t1);
  const float a0 = anchors[4 * g + 0], a1 = anchors[4 * g + 1];
  const float a2 = anchors[4 * g + 2], a3 = anchors[4 * g + 3];
  const float aw = a2 - a0, ah = a3 - a1;
  const float ax = a0 + 0.5f * aw, ay = a1 + 0.5f * ah;
  const float tw = t2 - t0, th = t3 - t1;
  const float tx = t0 + 0.5f * tw, ty = t1 + 0.5f * th;
  const float gp0 = (tx - ax) / aw, gp1 = (ty - ay) / ah;
  const float gp2 = logf(tw / aw),  gp3 = logf(th / ah);

  const int C  = (A + NBLK - 1) / NBLK;
  const int lo = blockIdx.x * C;
  const int hi = min(lo + C, A);

  float q_pos = 0.f, q_neg = 0.f, q_cls = 0.f, q_reg = 0.f;
  float best_iou = -1.0f;
  unsigned int best_idx = 0xFFFFFFFFu;

  const float4* __restrict__ reg4 = reinterpret_cast<const float4*>(regions);

  for (int a = lo + sub; a < hi; a += 4) {
    const float4 r = reg4[a];             // wave-uniform address -> scalar load
    float ix1 = fmaxf(t0, r.x), iy1 = fmaxf(t1, r.y);
    float ix2 = fminf(t2, r.z), iy2 = fminf(t3, r.w);
    float iw = fmaxf(ix2 - ix1, 0.f), ihh = fmaxf(iy2 - iy1, 0.f);
    float inter  = iw * ihh;
    float area_b = (r.z - r.x) * (r.w - r.y);
    float iou    = inter / (area_g + area_b - inter);
    if (iou > best_iou) { best_iou = iou; best_idx = (unsigned)a; }
    if (iou > C_IOU_POS) {
      q_pos += 1.0f;
      q_cls += 1.0f - scores[a] + C_LOG1PEXP;
      float rw = r.z - r.x, rh = r.w - r.y;
      float rx = r.x + 0.5f * rw, ry = r.y + 0.5f * rh;
      float d0 = (rx - ax) / aw - gp0;
      float d1 = (ry - ay) / ah - gp1;
      float d2 = logf(rw / aw) - gp2;
      float d3 = logf(rh / ah) - gp3;
      q_reg += 0.25f * (sl1f(d0) + sl1f(d1) + sl1f(d2) + sl1f(d3));
    } else if (iou < C_IOU_NEG) {
      q_neg += 1.0f;
    }
  }

  // per-g global argmax: key = iou bits (monotonic, iou>=0) | ~idx (tie -> first)
  if (best_idx != 0xFFFFFFFFu) {
    unsigned long long key =
        ((unsigned long long)__float_as_uint(best_iou) << 32) |
        (unsigned long long)(0xFFFFFFFFu - best_idx);
    atomicMax(&best_key[g], key);
  }

  // exact wave-level reduction of the 4 accumulators via v_wmma chain
  float tot = wave_reduce4(lane, q_pos, q_neg, q_cls, q_reg);

  __shared__ float red[8 * 4];
  if (lane < 4) red[wave * 4 + lane] = tot;
  __syncthreads();
  if (t == 0) {
    float p = 0.f, n = 0.f, c = 0.f, r = 0.f;
    for (int w = 0; w < 8; ++w) {
      p += red[w * 4 + 0];
      n += red[w * 4 + 1];
      c += red[w * 4 + 2];
      r += red[w * 4 + 3];
    }
    atomicAdd(&cnts[0], (int)(p + 0.5f));   // pos count (exact integer f32)
    atomicAdd(&cnts[1], (int)(n + 0.5f));   // neg count
    cls_partial[blockIdx.x] = c;            // deterministic partials
    reg_partial[blockIdx.x] = r;
  }
}

// ---------------- pass 2: label fix-up + final loss ----------------
__global__ void rpn_final(const float* __restrict__ scores,
                          const float* __restrict__ regions,
                          const float* __restrict__ anchors,
                          const float* __restrict__ gt,
                          const unsigned long long* __restrict__ best_key,
                          const int* __restrict__ cnts,
                          const float* __restrict__ cls_partial,
                          const float* __restrict__ reg_partial,
                          float* __restrict__ out) {
  const int g = threadIdx.x;  // 64 threads
  __shared__ float s_cls[GNUM], s_reg[GNUM];
  __shared__ int   s_dp[GNUM], s_dn[GNUM];

  // deterministic fixed-order sum of per-block partials
  float csum = 0.f, rsum = 0.f;
  for (int b = g; b < NBLK; b += GNUM) {
    csum += cls_partial[b];
    rsum += reg_partial[b];
  }

  unsigned long long key = best_key[g];
  unsigned int bidx = 0xFFFFFFFFu - (unsigned int)(key & 0xFFFFFFFFull);
  float biou = __uint_as_float((unsigned int)(key >> 32));

  int dp = 0, dn = 0;
  if (!(biou > C_IOU_POS)) {  // labels[g, best] forced 0/-1 -> 1
    dp = 1;
    csum += 1.0f - scores[bidx] + C_LOG1PEXP;
    const float4 r = reinterpret_cast<const float4*>(regions)[bidx];
    const float t0 = gt[4 * g + 0], t1 = gt[4 * g + 1];
    const float t2 = gt[4 * g + 2], t3 = gt[4 * g + 3];
    const float a0 = anchors[4 * g + 0], a1 = anchors[4 * g + 1];
    const float a2 = anchors[4 * g + 2], a3 = anchors[4 * g + 3];
    const float aw = a2 - a0, ah = a3 - a1;
    const float ax = a0 + 0.5f * aw, ay = a1 + 0.5f * ah;
    const float tw = t2 - t0, th = t3 - t1;
    const float tx = t0 + 0.5f * tw, ty = t1 + 0.5f * th;
    float rw = r.z - r.x, rh = r.w - r.y;
    float rx = r.x + 0.5f * rw, ry = r.y + 0.5f * rh;
    float d0 = (rx - ax) / aw - (tx - ax) / aw;
    float d1 = (ry - ay) / ah - (ty - ay) / ah;
    float d2 = logf(rw / aw) - logf(tw / aw);
    float d3 = logf(rh / ah) - logf(th / ah);
    rsum += 0.25f * (sl1f(d0) + sl1f(d1) + sl1f(d2) + sl1f(d3));
    if (biou < C_IOU_NEG) dn = -1;  // was counted negative, now positive
  }

  s_cls[g] = csum; s_reg[g] = rsum; s_dp[g] = dp; s_dn[g] = dn;
  if (g == GNUM - 1) out[1] = (float)bidx;  // best[-1]
  __syncthreads();

  if (g == 0) {
    float cs = 0.f, rs = 0.f;
    int P = cnts[0], N = cnts[1];
    for (int i = 0; i < GNUM; ++i) {
      cs += s_cls[i]; rs += s_reg[i]; P += s_dp[i]; N += s_dn[i];
    }
    float cls_sum = cs + (float)N * C_LN2;
    int   cls_cnt = P + N;
    float loss = cls_sum / (float)max(cls_cnt, 1) +
                 C_LAMBDA * rs / (float)max(P, 1);  // K_NORM == 1
    out[0] = loss;
  }
}

// ---------------- launcher ----------------
extern "C" void kernel_launch(void* const* d_in, const int* in_sizes, int n_in,
                              void* d_out, int out_size, void* d_ws,
                              size_t ws_size, hipStream_t stream) {
  (void)n_in; (void)out_size; (void)ws_size;
  const float* scores  = (const float*)d_in[0];
  const float* regions = (const float*)d_in[1];
  const float* anchors = (const float*)d_in[2];
  const float* gt      = (const float*)d_in[3];
  const int A = in_sizes[0];
  float* out = (float*)d_out;

  // workspace layout
  unsigned long long* best_key = (unsigned long long*)d_ws;          // 64 * u64
  int*   cnts        = (int*)((char*)d_ws + 512);                    // pos, neg
  float* cls_partial = (float*)((char*)d_ws + 768);                  // NBLK
  float* reg_partial = cls_partial + NBLK;                           // NBLK

  rpn_init<<<1, 64, 0, stream>>>(best_key, cnts);
  rpn_main<<<NBLK, TPB, 0, stream>>>(scores, regions, anchors, gt, best_key,
                                     cnts, cls_partial, reg_partial, A);
  rpn_final<<<1, GNUM, 0, stream>>>(scores, regions, anchors, gt, best_key,
                                    cnts, cls_partial, reg_partial, out);
}